// EdgeSelector_37082747634231
// MI455X (gfx1250) — compile-verified
//
#include <hip/hip_runtime.h>
#include <hip/hip_bf16.h>

typedef __attribute__((ext_vector_type(2))) float v2f;
typedef __attribute__((ext_vector_type(8))) float v8f;
typedef __attribute__((ext_vector_type(4))) unsigned int v4u;
typedef __attribute__((ext_vector_type(4))) int v4i;
typedef __attribute__((ext_vector_type(8))) int v8i;

#define DD 256   // in_dim
#define HH 256   // hidden
#define ENSZ 2   // ensemble

#define WMMA_F32(a, b, c) \
  __builtin_amdgcn_wmma_f32_16x16x4_f32(false, (a), false, (b), (short)0, (c), false, false)

// ---------------------------------------------------------------------------
// TDM: DMA one 32-row x 256-col fp32 tile (32KB, contiguous rows) from global
// memory into LDS. D# packing per CDNA5 ISA:
//  group0: count=1 | lds_addr | global_addr[56:0] | type=2
//  group1: data_size=4B, tensor_dim0=256, tensor_dim1=256, tile=256x32,
//          dim0_stride=256, dim1_stride=65536
// Issued once per wave (EXEC-independent); tracked by TENSORcnt.
// ---------------------------------------------------------------------------
__device__ __forceinline__ void tdm_load_tile32x256(const float* gsrc,
                                                    unsigned int lds_off)
{
  unsigned long long ga = (unsigned long long)(uintptr_t)gsrc;
  v4u g0;
  g0.x = 1u;                                               // count=1
  g0.y = lds_off;                                          // lds_addr (bytes)
  g0.z = (unsigned int)ga;                                 // global_addr[31:0]
  g0.w = ((unsigned int)(ga >> 32) & 0x01FFFFFFu) | (2u << 30); // [56:32]|type=2
  v8i g1;
  g1[0] = 0x00020000;   // data_size=2 (4 bytes), mask=0
  g1[1] = 0x01000000;   // tensor_dim0[15:0]=256 in bits[31:16]
  g1[2] = 0x01000000;   // tensor_dim1[15:0]=256 in bits[31:16]
  g1[3] = 0x01000000;   // tile_dim0=256 in bits[31:16]
  g1[4] = 32;           // tile_dim1=32, tile_dim2=0
  g1[5] = 256;          // tensor_dim0_stride[31:0]=256
  g1[6] = 0;            // stride0[47:32]=0, stride1[15:0]=0
  g1[7] = 1;            // tensor_dim1_stride = 65536 (bits[47:16]=1)
  v4i z4 = {0, 0, 0, 0};
#if defined(__clang_major__) && (__clang_major__ >= 23)
  v8i z8 = {0, 0, 0, 0, 0, 0, 0, 0};
  __builtin_amdgcn_tensor_load_to_lds(g0, g1, z4, z4, z8, 0);
#else
  __builtin_amdgcn_tensor_load_to_lds(g0, g1, z4, z4, 0);
#endif
}

// ---------------------------------------------------------------------------
// Fused node kernel:  x = node_feat @ W_enc + b_enc        (x stays in LDS)
//                     P_top = x @ W1[0:256,:]   + b1       (folded once)
//                     P_bot = x @ W1[256:512,:]
// Block = 256 threads (8 wave32), node tile M = 32.
// ---------------------------------------------------------------------------
__global__ __launch_bounds__(256) void node_kernel(
    const float* __restrict__ NF, const float* __restrict__ Wenc,
    const float* __restrict__ benc, const float* __restrict__ W1,
    const float* __restrict__ b1v,
    float* __restrict__ Ptop, float* __restrict__ Pbot, int Nrows)
{
  __shared__ float Anf[32 * DD];   // node_feat tile   (32KB)
  __shared__ float Xs[32 * DD];    // encoded tile     (32KB)

  const int tid = threadIdx.x;
  const int m0 = blockIdx.x * 32;

  {
    float4* d4 = (float4*)Anf;
#pragma unroll
    for (int r = 0; r < 8; ++r) {
      int i4 = tid + r * 256;
      int row = i4 >> 6, c4 = i4 & 63;
      int rr = m0 + row; if (rr >= Nrows) rr = Nrows - 1;
      d4[i4] = ((const float4*)(NF + (size_t)rr * DD))[c4];
    }
  }
  __syncthreads();

  const int lane = tid & 31;
  const int w = tid >> 5;
  const int m = lane & 15;
  const int hi = lane >> 4;
  const int n0a = w * 16;
  const int n0b = (w + 8) * 16;
  const int na = n0a + m, nb = n0b + m;

  // ---- Encoder GEMM -------------------------------------------------------
  {
    v8f c00 = {}, c01 = {}, c10 = {}, c11 = {};
#pragma unroll 2
    for (int k = 0; k < DD; k += 4) {
      int ka = k + 2 * hi;
      v2f a0 = *(const v2f*)(Anf + m * DD + ka);
      v2f a1 = *(const v2f*)(Anf + (m + 16) * DD + ka);
      v2f bA, bB;
      bA.x = Wenc[(size_t)ka * DD + na];
      bA.y = Wenc[(size_t)(ka + 1) * DD + na];
      bB.x = Wenc[(size_t)ka * DD + nb];
      bB.y = Wenc[(size_t)(ka + 1) * DD + nb];
      c00 = WMMA_F32(a0, bA, c00);
      c10 = WMMA_F32(a1, bA, c10);
      c01 = WMMA_F32(a0, bB, c01);
      c11 = WMMA_F32(a1, bB, c11);
    }
    float ba = benc[na], bb = benc[nb];
#pragma unroll
    for (int j = 0; j < 8; ++j) {
      int mm = j + 8 * hi;
      Xs[mm * DD + na]        = c00[j] + ba;
      Xs[(mm + 16) * DD + na] = c10[j] + ba;
      Xs[mm * DD + nb]        = c01[j] + bb;
      Xs[(mm + 16) * DD + nb] = c11[j] + bb;
    }
  }
  __syncthreads();

  // ---- Layer-1 partials ---------------------------------------------------
  {
    const float* W1t = W1;
    const float* W1b = W1 + (size_t)DD * HH;
    v8f t00 = {}, t01 = {}, t10 = {}, t11 = {};
    v8f u00 = {}, u01 = {}, u10 = {}, u11 = {};
#pragma unroll 2
    for (int k = 0; k < DD; k += 4) {
      int ka = k + 2 * hi;
      v2f a0 = *(const v2f*)(Xs + m * DD + ka);
      v2f a1 = *(const v2f*)(Xs + (m + 16) * DD + ka);
      v2f bAt, bBt, bAb, bBb;
      bAt.x = W1t[(size_t)ka * HH + na];
      bAt.y = W1t[(size_t)(ka + 1) * HH + na];
      bBt.x = W1t[(size_t)ka * HH + nb];
      bBt.y = W1t[(size_t)(ka + 1) * HH + nb];
      bAb.x = W1b[(size_t)ka * HH + na];
      bAb.y = W1b[(size_t)(ka + 1) * HH + na];
      bBb.x = W1b[(size_t)ka * HH + nb];
      bBb.y = W1b[(size_t)(ka + 1) * HH + nb];
      t00 = WMMA_F32(a0, bAt, t00);
      t10 = WMMA_F32(a1, bAt, t10);
      t01 = WMMA_F32(a0, bBt, t01);
      t11 = WMMA_F32(a1, bBt, t11);
      u00 = WMMA_F32(a0, bAb, u00);
      u10 = WMMA_F32(a1, bAb, u10);
      u01 = WMMA_F32(a0, bBb, u01);
      u11 = WMMA_F32(a1, bBb, u11);
    }
    float ba = b1v[na], bb = b1v[nb];  // b1 folded into Ptop only
#pragma unroll
    for (int j = 0; j < 8; ++j) {
      int mm = j + 8 * hi;
      int r0 = m0 + mm, r1 = m0 + mm + 16;
      if (r0 < Nrows) {
        Ptop[(size_t)r0 * HH + na] = t00[j] + ba;
        Ptop[(size_t)r0 * HH + nb] = t01[j] + bb;
        Pbot[(size_t)r0 * HH + na] = u00[j];
        Pbot[(size_t)r0 * HH + nb] = u01[j];
      }
      if (r1 < Nrows) {
        Ptop[(size_t)r1 * HH + na] = t10[j] + ba;
        Ptop[(size_t)r1 * HH + nb] = t11[j] + bb;
        Pbot[(size_t)r1 * HH + na] = u10[j];
        Pbot[(size_t)r1 * HH + nb] = u11[j];
      }
    }
  }
}

// ---------------------------------------------------------------------------
__global__ void prefix_kernel(const int* __restrict__ nnodes,
                              const int* __restrict__ necand, int G,
                              int* __restrict__ cum_nodes,
                              int* __restrict__ cum_edges)
{
  if (threadIdx.x == 0 && blockIdx.x == 0) {
    int cn = 0, ce = 0;
    for (int g = 0; g < G; ++g) {
      cum_nodes[g] = cn; cum_edges[g] = ce;
      cn += nnodes[g];  ce += necand[g];
    }
    cum_edges[G] = ce;
  }
}

// ---------------------------------------------------------------------------
__global__ __launch_bounds__(256) void edge_idx_kernel(
    const int* __restrict__ edge_candidate,
    const int* __restrict__ cum_nodes, const int* __restrict__ cum_edges,
    int G, int E, int* __restrict__ idx, float* __restrict__ idx_f)
{
  int e = blockIdx.x * blockDim.x + threadIdx.x;
  if (e >= E) return;
  int lo = 0, hiq = G;
  while (hiq - lo > 1) {
    int mid = (lo + hiq) >> 1;
    if (cum_edges[mid] <= e) lo = mid; else hiq = mid;
  }
  int off = cum_nodes[lo];
  int s = edge_candidate[2 * e + 0] + off;
  int d = edge_candidate[2 * e + 1] + off;
  idx[2 * e + 0] = s;       idx[2 * e + 1] = d;
  idx_f[2 * e + 0] = (float)s;
  idx_f[2 * e + 1] = (float)d;
}

// ---------------------------------------------------------------------------
// Edge kernel. Block = 256 threads (8 wave32), 32 edges per block.
//   H1 [32,256] = relu(Ptop[src] + Pbot[dst])           (b1 pre-folded)
//   H2 [32,256] = relu(H1 @ W2 + b2)   -- W2 streamed via double-buffered TDM
//   scores [32,2] = H2 @ W3 + b3
// ---------------------------------------------------------------------------
__global__ __launch_bounds__(256) void edge_kernel(
    const float* __restrict__ Ptop, const float* __restrict__ Pbot,
    const int* __restrict__ idx,
    const float* __restrict__ W2, const float* __restrict__ b2,
    const float* __restrict__ W3, const float* __restrict__ b3,
    float* __restrict__ scores, int E)
{
  __shared__ float Hs[32 * HH];       // H1, later H2 (32KB)
  __shared__ float Wb[2][32 * HH];    // double-buffered W2 chunks (2x32KB)

  const int tid = threadIdx.x;
  const int e0 = blockIdx.x * 32;
  const int wave = tid >> 5;

  // Gather + add + ReLU: 8 threads per edge, 32 floats (8 float4) each.
  {
    int i = tid >> 3;
    int col0 = (tid & 7) * 32;
    int ei = e0 + i; if (ei >= E) ei = E - 1;
    int s = idx[(size_t)ei * 2 + 0];
    int d = idx[(size_t)ei * 2 + 1];
    const float4* ps = (const float4*)(Ptop + (size_t)s * HH + col0);
    const float4* pd = (const float4*)(Pbot + (size_t)d * HH + col0);
    float4* o4 = (float4*)(Hs + i * HH + col0);
#pragma unroll
    for (int r = 0; r < 8; ++r) {
      float4 a = ps[r], b = pd[r];
      float4 v;
      v.x = fmaxf(a.x + b.x, 0.0f);
      v.y = fmaxf(a.y + b.y, 0.0f);
      v.z = fmaxf(a.z + b.z, 0.0f);
      v.w = fmaxf(a.w + b.w, 0.0f);
      o4[r] = v;
    }
  }

  const unsigned int lds0 = (unsigned int)(uintptr_t)(&Wb[0][0]);
  const unsigned int lds1 = (unsigned int)(uintptr_t)(&Wb[1][0]);

  // Prefetch W2 chunk 0 via TDM (wave 0 only; per-wave TENSORcnt).
  if (wave == 0) {
    tdm_load_tile32x256(W2, lds0);
    __builtin_amdgcn_s_wait_tensorcnt(0);
  }
  __syncthreads();   // gather done + chunk 0 resident

  const int lane = tid & 31;
  const int m = lane & 15;
  const int hi = lane >> 4;
  const int n0a = wave * 16;
  const int n0b = (wave + 8) * 16;
  const int na = n0a + m, nb = n0b + m;

  // GEMM2: [32,256] @ [256,256], K chunked by 32 with TDM double buffering.
  v8f c00 = {}, c01 = {}, c10 = {}, c11 = {};
  for (int c = 0; c < HH / 32; ++c) {
    if (wave == 0 && (c + 1) < HH / 32)
      tdm_load_tile32x256(W2 + (size_t)(c + 1) * 32 * HH,
                          (c & 1) ? lds0 : lds1);
    const float* Wc = Wb[c & 1];
    const float* Ac = Hs + c * 32;
#pragma unroll 2
    for (int kk = 0; kk < 32; kk += 4) {
      int ka = kk + 2 * hi;
      v2f a0 = *(const v2f*)(Ac + m * HH + ka);
      v2f a1 = *(const v2f*)(Ac + (m + 16) * HH + ka);
      v2f bA, bB;
      bA.x = Wc[ka * HH + na];
      bA.y = Wc[(ka + 1) * HH + na];
      bB.x = Wc[ka * HH + nb];
      bB.y = Wc[(ka + 1) * HH + nb];
      c00 = WMMA_F32(a0, bA, c00);
      c10 = WMMA_F32(a1, bA, c10);
      c01 = WMMA_F32(a0, bB, c01);
      c11 = WMMA_F32(a1, bB, c11);
    }
    if (wave == 0) __builtin_amdgcn_s_wait_tensorcnt(0);
    __syncthreads();   // next chunk resident; this chunk's reads complete
  }

  // H2 = relu(acc + b2), overwriting Hs (all H1 reads completed above).
  {
    float ba = b2[na], bb = b2[nb];
#pragma unroll
    for (int j = 0; j < 8; ++j) {
      int mm = j + 8 * hi;
      Hs[mm * HH + na]        = fmaxf(c00[j] + ba, 0.0f);
      Hs[(mm + 16) * HH + na] = fmaxf(c10[j] + ba, 0.0f);
      Hs[mm * HH + nb]        = fmaxf(c01[j] + bb, 0.0f);
      Hs[(mm + 16) * HH + nb] = fmaxf(c11[j] + bb, 0.0f);
    }
  }
  __syncthreads();

  // GEMM3: [32,256] @ [256,2] + b3 (tiny N -> VALU dots)
  if (tid < 32 * ENSZ) {
    int i = tid >> 1, en = tid & 1;
    float s = b3[en];
#pragma unroll 8
    for (int k = 0; k < HH; ++k)
      s = fmaf(Hs[i * HH + k], W3[(size_t)k * ENSZ + en], s);
    if (e0 + i < E) scores[(size_t)(e0 + i) * ENSZ + en] = s;
  }
}

// ---------------------------------------------------------------------------
extern "C" void kernel_launch(void* const* d_in, const int* in_sizes, int n_in,
                              void* d_out, int out_size, void* d_ws, size_t ws_size,
                              hipStream_t stream)
{
  const float* node_feat      = (const float*)d_in[0];
  const int*   edge_candidate = (const int*)d_in[1];
  const int*   nnodes         = (const int*)d_in[2];
  const int*   necand         = (const int*)d_in[3];
  const float* W_enc          = (const float*)d_in[4];
  const float* b_enc          = (const float*)d_in[5];
  const float* W1             = (const float*)d_in[6];
  const float* b1             = (const float*)d_in[7];
  const float* W2             = (const float*)d_in[8];
  const float* b2             = (const float*)d_in[9];
  const float* W3             = (const float*)d_in[10];
  const float* b3             = (const float*)d_in[11];

  const int N = in_sizes[0] / DD;
  const int E = in_sizes[1] / 2;
  const int G = in_sizes[2];

  float* Ptop      = (float*)d_ws;
  float* Pbot      = Ptop + (size_t)N * HH;
  int*   idx       = (int*)(Pbot + (size_t)N * HH);
  int*   cum_nodes = idx + (size_t)2 * E;
  int*   cum_edges = cum_nodes + G;

  float* scores = (float*)d_out;
  float* idx_f  = scores + (size_t)2 * E;

  node_kernel<<<(N + 31) / 32, 256, 0, stream>>>(node_feat, W_enc, b_enc,
                                                 W1, b1, Ptop, Pbot, N);
  prefix_kernel<<<1, 32, 0, stream>>>(nnodes, necand, G, cum_nodes, cum_edges);
  edge_idx_kernel<<<(E + 255) / 256, 256, 0, stream>>>(edge_candidate, cum_nodes,
                                                       cum_edges, G, E, idx, idx_f);
  edge_kernel<<<(E + 31) / 32, 256, 0, stream>>>(Ptop, Pbot, idx, W2, b2, W3, b3,
                                                 scores, E);
}